// SwitchGate_5033701671493
// MI455X (gfx1250) — compile-verified
//
#include <hip/hip_runtime.h>
#include <math.h>

typedef __attribute__((ext_vector_type(2))) float v2f;
typedef __attribute__((ext_vector_type(8))) float v8f;

#define NUM_EXPERTS 64
#define ROWS_PER_BLOCK 1024
#define THREADS 256
#define WAVES 8
#define ROW_ITERS (ROWS_PER_BLOCK / (WAVES * 2)) // 64
#define FIN_THREADS 512                           // 16 waves: 4 expert-groups x 4 k-stripes

// ---------------------------------------------------------------------------
// Main streaming kernel: one pass over logits, everything in registers.
// Half-wave (16 lanes) per row, float4 per lane -> one coalesced b128/row-half.
// Workspace partials are written EXPERT-MAJOR: ws[e * nblk_pad + block].
// ---------------------------------------------------------------------------
__global__ __launch_bounds__(THREADS) void switchgate_main(
    const float* __restrict__ logits, float* __restrict__ out,
    float* __restrict__ ws_sums, float* __restrict__ ws_counts,
    int Bn, int nblk_pad)
{
    __shared__ float    s_acc[WAVES * NUM_EXPERTS];
    __shared__ unsigned s_count[NUM_EXPERTS];
    __shared__ int      s_idx[ROWS_PER_BLOCK];
    __shared__ float    s_pmax[ROWS_PER_BLOCK];

    const int tid  = threadIdx.x;
    const int wave = tid >> 5;
    const int lane = tid & 31;
    const int half = lane >> 4;   // 0: row 2w, 1: row 2w+1
    const int sub  = lane & 15;   // lane within half: owns experts 4*sub..4*sub+3
    const int rowBase = (int)blockIdx.x * ROWS_PER_BLOCK;

    if (tid < NUM_EXPERTS) s_count[tid] = 0u;
    __syncthreads();

    float acc0 = 0.f, acc1 = 0.f, acc2 = 0.f, acc3 = 0.f;
    const float LOG2E = 1.4426950408889634f;

    for (int it = 0; it < ROW_ITERS; ++it) {
        const int r = rowBase + it * (WAVES * 2) + wave * 2 + half;
        if (r < Bn) { // uniform across the 16-lane half -> shuffles safe
            const float4* p =
                reinterpret_cast<const float4*>(logits + (size_t)r * NUM_EXPERTS) + sub;
            // stream-prefetch 8 iterations ahead (global_prefetch_b8)
            const int rp = r + 8 * (WAVES * 2);
            if (rp < Bn)
                __builtin_prefetch(
                    reinterpret_cast<const float4*>(logits + (size_t)rp * NUM_EXPERTS) + sub,
                    0, 1);

            const float4 x = *p;

            // local argmax (first index wins on ties, like jnp.argmax)
            float m = x.x; int mi = sub * 4;
            if (x.y > m) { m = x.y; mi = sub * 4 + 1; }
            if (x.z > m) { m = x.z; mi = sub * 4 + 2; }
            if (x.w > m) { m = x.w; mi = sub * 4 + 3; }
            #pragma unroll
            for (int mask = 1; mask <= 8; mask <<= 1) {
                float om  = __shfl_xor(m, mask, 32);
                int   omi = __shfl_xor(mi, mask, 32);
                if (om > m || (om == m && omi < mi)) { m = om; mi = omi; }
            }

            const float e0 = __builtin_amdgcn_exp2f((x.x - m) * LOG2E);
            const float e1 = __builtin_amdgcn_exp2f((x.y - m) * LOG2E);
            const float e2 = __builtin_amdgcn_exp2f((x.z - m) * LOG2E);
            const float e3 = __builtin_amdgcn_exp2f((x.w - m) * LOG2E);
            float s = (e0 + e1) + (e2 + e3);
            #pragma unroll
            for (int mask = 1; mask <= 8; mask <<= 1) s += __shfl_xor(s, mask, 32);

            const float inv = 1.0f / s; // max prob == exp(0)/s == 1/s
            acc0 += e0 * inv; acc1 += e1 * inv; acc2 += e2 * inv; acc3 += e3 * inv;

            if (sub == 0) {
                const int slot = it * (WAVES * 2) + wave * 2 + half;
                s_idx[slot]  = mi;
                s_pmax[slot] = inv;
                atomicAdd(&s_count[mi], 1u); // integer LDS atomic: deterministic
            }
        }
    }

    // combine the two half-wave copies of the expert accumulators
    acc0 += __shfl_xor(acc0, 16, 32);
    acc1 += __shfl_xor(acc1, 16, 32);
    acc2 += __shfl_xor(acc2, 16, 32);
    acc3 += __shfl_xor(acc3, 16, 32);
    if (half == 0) {
        s_acc[wave * NUM_EXPERTS + sub * 4 + 0] = acc0;
        s_acc[wave * NUM_EXPERTS + sub * 4 + 1] = acc1;
        s_acc[wave * NUM_EXPERTS + sub * 4 + 2] = acc2;
        s_acc[wave * NUM_EXPERTS + sub * 4 + 3] = acc3;
    }
    __syncthreads();

    // deterministic fixed-order reduce across the 8 waves -> per-block partials
    if (tid < NUM_EXPERTS) {
        float t = 0.f;
        #pragma unroll
        for (int w = 0; w < WAVES; ++w) t += s_acc[w * NUM_EXPERTS + tid];
        ws_sums[(size_t)tid * nblk_pad + blockIdx.x]   = t;                   // expert-major
        ws_counts[(size_t)tid * nblk_pad + blockIdx.x] = (float)s_count[tid]; // expert-major
    }

    // coalesced per-row outputs: idx [0,B), multiplier [B,2B), pmax [2B+1,3B+1)
    const size_t Bs = (size_t)Bn;
    for (int rloc = tid; rloc < ROWS_PER_BLOCK; rloc += THREADS) {
        const int row = rowBase + rloc;
        if (row < Bn) {
            const float pm = s_pmax[rloc];
            out[(size_t)row]              = (float)s_idx[rloc]; // route_idx
            out[Bs + (size_t)row]         = pm;                 // route_multiplier
            out[2 * Bs + 1 + (size_t)row] = pm;                 // route_prob_max
        }
    }
}

// ---------------------------------------------------------------------------
// Zero only the pad slots [nblk, nblk_pad) of the expert-major workspace so
// the finalize loop needs no bounds checks. Launched only when pad > 0.
// ---------------------------------------------------------------------------
__global__ void switchgate_zero_pad(float* __restrict__ ws_sums,
                                    float* __restrict__ ws_counts,
                                    int nblk, int nblk_pad)
{
    const int pad = nblk_pad - nblk;
    const int total = NUM_EXPERTS * pad;
    const int i = (int)blockIdx.x * (int)blockDim.x + (int)threadIdx.x;
    if (i < total) {
        const int e = i / pad;
        const int p = nblk + (i % pad);
        ws_sums[(size_t)e * nblk_pad + p]   = 0.f;
        ws_counts[(size_t)e * nblk_pad + p] = 0.f;
    }
}

// ---------------------------------------------------------------------------
// Finalize: column-sum of per-block partials via chained V_WMMA_F32_16X16X4_F32
// with an all-ones A (D[m,n] = C[m,n] + sum_k B[k,n]). Expert-major layout
// gives each lane contiguous, unconditional global_load_b64 operands:
// lane (n, khalf) supplies K = {4j + 2*khalf, 4j + 2*khalf + 1} of column n.
// Loads are batched into distinct registers (16 in flight) BEFORE the WMMA
// chain so only one L2 latency is exposed per 32 K-values.
// 16 waves: 4 expert-groups x 4 K-stripes; stripes combined in fixed order.
// ---------------------------------------------------------------------------
__global__ __launch_bounds__(FIN_THREADS) void switchgate_finalize(
    const float* __restrict__ wsS, const float* __restrict__ wsC,
    float* __restrict__ out, int nblk_pad, int Bn)
{
    __shared__ float s_psum[4 * NUM_EXPERTS];
    __shared__ float s_pcnt[4 * NUM_EXPERTS];
    __shared__ float s_mean[NUM_EXPERTS];
    __shared__ float s_frac[NUM_EXPERTS];

    const int tid    = threadIdx.x;
    const int wv     = tid >> 5;       // 0..15
    const int lane   = tid & 31;
    const int n      = lane & 15;      // D column -> expert within group
    const int khalf  = lane >> 4;
    const int grp    = wv & 3;         // expert group: experts 16*grp..16*grp+15
    const int stripe = wv >> 2;        // K stripe 0..3
    const int e      = grp * 16 + n;

    const int chunk = nblk_pad >> 2;   // nblk_pad % 128 == 0 -> chunk % 32 == 0
    const int pbase = stripe * chunk;

    v2f a = {1.0f, 1.0f};              // all-ones A (16x4)
    v8f cs = {};                       // expert prob-sum accumulator
    v8f cc = {};                       // expert count accumulator

    const float* rowS = wsS + (size_t)e * nblk_pad + pbase + khalf * 2;
    const float* rowC = wsC + (size_t)e * nblk_pad + pbase + khalf * 2;

    for (int p = 0; p < chunk; p += 32) {
        v2f bs[8], bc[8];
        #pragma unroll
        for (int j = 0; j < 8; ++j) {        // 16 independent b64 loads in flight
            bs[j] = *(const v2f*)(rowS + p + 4 * j);
            bc[j] = *(const v2f*)(rowC + p + 4 * j);
        }
        #pragma unroll
        for (int j = 0; j < 8; ++j) {        // then the accumulate chain
            cs = __builtin_amdgcn_wmma_f32_16x16x4_f32(false, a, false, bs[j], (short)0, cs, false, false);
            cc = __builtin_amdgcn_wmma_f32_16x16x4_f32(false, a, false, bc[j], (short)0, cc, false, false);
        }
    }

    if (lane < 16) {                   // lanes 0-15 hold row M=0 of D (col n)
        s_psum[stripe * NUM_EXPERTS + e] = cs[0];
        s_pcnt[stripe * NUM_EXPERTS + e] = cc[0];
    }
    __syncthreads();

    if (tid < NUM_EXPERTS) {           // fixed-order stripe combine: deterministic
        float ts = 0.f, tc = 0.f;
        #pragma unroll
        for (int s2 = 0; s2 < 4; ++s2) {
            ts += s_psum[s2 * NUM_EXPERTS + tid];
            tc += s_pcnt[s2 * NUM_EXPERTS + tid];
        }
        const float invB = 1.0f / (float)Bn;
        s_mean[tid] = ts * invB;       // route_prob_mean
        s_frac[tid] = tc * invB;       // route_frac
    }
    __syncthreads();

    if (tid == 0) {
        float dot = 0.f, fmin = s_frac[0], fmax = s_frac[0], fsum = 0.f;
        for (int i = 0; i < NUM_EXPERTS; ++i) {
            const float f = s_frac[i];
            dot += f * s_mean[i];
            fmin = (f < fmin) ? f : fmin;
            fmax = (f > fmax) ? f : fmax;
            fsum += f;
        }
        const float fmean = fsum / (float)NUM_EXPERTS;
        float var = 0.f;
        for (int i = 0; i < NUM_EXPERTS; ++i) {
            const float d = s_frac[i] - fmean;
            var += d * d;
        }
        var /= (float)(NUM_EXPERTS - 1);  // ddof=1 (torch.std default)
        const size_t Bs = (size_t)Bn;
        out[2 * Bs]     = (float)NUM_EXPERTS * dot - 1.0f; // load_balancing_loss
        out[3 * Bs + 1] = fmin;
        out[3 * Bs + 2] = fmax;
        out[3 * Bs + 3] = sqrtf(var);
    }
}

extern "C" void kernel_launch(void* const* d_in, const int* in_sizes, int n_in,
                              void* d_out, int out_size, void* d_ws, size_t ws_size,
                              hipStream_t stream) {
    (void)n_in; (void)out_size; (void)ws_size;
    const float* logits = (const float*)d_in[0];
    const int total    = in_sizes[0];
    const int Bn       = total / NUM_EXPERTS;
    const int nblk     = (Bn + ROWS_PER_BLOCK - 1) / ROWS_PER_BLOCK;
    const int nblk_pad = (nblk + 127) & ~127;  // multiple of 128 (=> chunk % 32 == 0)

    float* ws_sums   = (float*)d_ws;                              // [64 * nblk_pad]
    float* ws_counts = ws_sums + (size_t)NUM_EXPERTS * nblk_pad;  // [64 * nblk_pad]
    float* out = (float*)d_out;

    if (nblk_pad > nblk) {
        const int padElems = NUM_EXPERTS * (nblk_pad - nblk);
        switchgate_zero_pad<<<(padElems + 255) / 256, 256, 0, stream>>>(
            ws_sums, ws_counts, nblk, nblk_pad);
    }
    switchgate_main<<<nblk, THREADS, 0, stream>>>(logits, out, ws_sums, ws_counts,
                                                  Bn, nblk_pad);
    switchgate_finalize<<<1, FIN_THREADS, 0, stream>>>(ws_sums, ws_counts, out,
                                                       nblk_pad, Bn);
}